// CombinedRegistrationLoss_14345190769122
// MI455X (gfx1250) — compile-verified
//
#include <hip/hip_runtime.h>
#include <math.h>

typedef float v2f __attribute__((ext_vector_type(2)));
typedef float v8f __attribute__((ext_vector_type(8)));

#define BATCH 8
#define NPTS  4096
#define TILES (NPTS / 16)        // 256 column tiles / row tiles
#define NPROB 4                  // 4 row-min problems (2 chamfers x 2 directions)

// ---------------------------------------------------------------------------
// Zero the 32 per-(problem,batch) accumulators. Must run every launch.
// ---------------------------------------------------------------------------
__global__ void zero_acc_kernel(float* __restrict__ acc) {
    if (threadIdx.x < NPROB * BATCH) acc[threadIdx.x] = 0.0f;
}

// ---------------------------------------------------------------------------
// Prep: apply pred/gt transforms to source points, pack all three point sets
// as float4 (v0, v1, v2, |v|^2) for the WMMA kernel.
// ---------------------------------------------------------------------------
__global__ void prep_points_kernel(const float* __restrict__ predT,
                                   const float* __restrict__ gtT,
                                   const float* __restrict__ src,
                                   const float* __restrict__ tgt,
                                   float4* __restrict__ packPred,
                                   float4* __restrict__ packTgt,
                                   float4* __restrict__ packGt) {
    int i = blockIdx.x * blockDim.x + threadIdx.x;     // 0 .. B*N-1
    if (i >= BATCH * NPTS) return;
    int b = i / NPTS;

    const float* sp = src + (size_t)i * 3;
    float h0 = sp[0], h1 = sp[1], h2 = sp[2];

    const float* P = predT + b * 16;
    const float* G = gtT + b * 16;
    // einsum('bnk,bjk->bnj'): row j of transform dotted with homogeneous point
    float p0 = P[0] * h0 + P[1] * h1 + P[2]  * h2 + P[3];
    float p1 = P[4] * h0 + P[5] * h1 + P[6]  * h2 + P[7];
    float p2 = P[8] * h0 + P[9] * h1 + P[10] * h2 + P[11];
    float g0 = G[0] * h0 + G[1] * h1 + G[2]  * h2 + G[3];
    float g1 = G[4] * h0 + G[5] * h1 + G[6]  * h2 + G[7];
    float g2 = G[8] * h0 + G[9] * h1 + G[10] * h2 + G[11];

    const float* tp = tgt + (size_t)i * 3;
    float t0 = tp[0], t1 = tp[1], t2 = tp[2];

    packPred[i] = make_float4(p0, p1, p2, p0 * p0 + p1 * p1 + p2 * p2);
    packGt[i]   = make_float4(g0, g1, g2, g0 * g0 + g1 * g1 + g2 * g2);
    packTgt[i]  = make_float4(t0, t1, t2, t0 * t0 + t1 * t1 + t2 * t2);
}

// ---------------------------------------------------------------------------
// Chamfer row-min kernel. One wave32 owns one 16-row tile of X for one
// (problem, batch) and sweeps all 256 column tiles of Y with
// V_WMMA_F32_16X16X4_F32:
//   A row (K=4)  = ( x0,  x1,  x2,  1   )
//   B col (K=4)  = (-2y0,-2y1,-2y2,|y|^2)
//   D[n,m]       = |y_m|^2 - 2 x_n.y_m        (add |x_n|^2 after the min)
//
// f32 A/B layout (16x4 / 4x16 over 32 lanes, 2 VGPRs): lanes 0-15 hold
// K={0,1}, lanes 16-31 hold K={2,3}, point index = lane%16.
// ---------------------------------------------------------------------------
__global__ void chamfer_rowmin_kernel(const float4* __restrict__ packPred,
                                      const float4* __restrict__ packTgt,
                                      const float4* __restrict__ packGt,
                                      float* __restrict__ acc) {
    int wid  = (blockIdx.x * blockDim.x + threadIdx.x) >> 5;
    int lane = threadIdx.x & 31;
    if (wid >= NPROB * BATCH * TILES) return;      // uniform per wave

    int p  = wid / (BATCH * TILES);
    int b  = (wid / TILES) % BATCH;
    int rt = wid % TILES;

    const float4* X;
    const float4* Y;
    switch (p) {
        case 0:  X = packPred; Y = packTgt;  break;  // per-pred nearest target
        case 1:  X = packTgt;  Y = packPred; break;  // per-target nearest pred
        case 2:  X = packPred; Y = packGt;   break;  // per-pred nearest gt
        default: X = packGt;   Y = packPred; break;  // per-gt nearest pred
    }

    int nloc = lane & 15;      // point within 16-tile
    int kp   = lane >> 4;      // 0 -> K{0,1}, 1 -> K{2,3}

    // A operand (hoisted, constant over the column sweep)
    const float* xr = (const float*)&X[(size_t)b * NPTS + rt * 16 + nloc];
    v2f a;
    a.x = xr[2 * kp];                       // x0 or x2
    a.y = kp ? 1.0f : xr[1];                // x1 or the K=3 pad value 1
    float x2 = xr[3];                       // |x_{nloc}|^2 (per-row constant)

    float selY = kp ? 1.0f : -2.0f;         // keep |y|^2 in the K=3 slot

    v8f czero = {};
    v8f runmin;
#pragma unroll
    for (int i = 0; i < 8; ++i) runmin[i] = 3.0e38f;

    const float* ybase = (const float*)&Y[(size_t)b * NPTS];
    for (int ct = 0; ct < TILES; ++ct) {
        const float* yr = ybase + (size_t)(ct * 16 + nloc) * 4 + 2 * kp;
        v2f bb;
        bb.x = -2.0f * yr[0];
        bb.y = selY  * yr[1];
        // D = A x B + 0   (v_wmma_f32_16x16x4_f32)
        v8f d = __builtin_amdgcn_wmma_f32_16x16x4_f32(
            /*neg_a=*/false, a, /*neg_b=*/false, bb,
            /*c_mod=*/(short)0, czero, /*reuse_a=*/false, /*reuse_b=*/false);
#pragma unroll
        for (int i = 0; i < 8; ++i) runmin[i] = fminf(runmin[i], d[i]);
    }

    // Min across the 16 columns held by each half-wave (C layout: VGPR i is
    // row M=i for lanes 0-15 and row M=8+i for lanes 16-31, N = lane%16).
#pragma unroll
    for (int m = 1; m <= 8; m <<= 1) {
#pragma unroll
        for (int i = 0; i < 8; ++i) {
            float o = __shfl_xor(runmin[i], m, 32);
            runmin[i] = fminf(runmin[i], o);
        }
    }

    // Sum of row-mins (D part): lower half holds rows 0..7, upper rows 8..15.
    float t = 0.0f;
#pragma unroll
    for (int i = 0; i < 8; ++i) t += runmin[i];
    t += __shfl_xor(t, 16, 32);             // all 16 rows

    // Sum of |x_r|^2 over the 16 rows (both halves compute the same value).
    float s2 = x2;
#pragma unroll
    for (int m = 1; m <= 8; m <<= 1) s2 += __shfl_xor(s2, m, 32);

    float wave_total = t + s2;              // sum over 16 rows of min distance
    if (lane == 0) atomicAdd(&acc[p * BATCH + b], wave_total);
}

// ---------------------------------------------------------------------------
// Finalize: transform losses + combine the four accumulated chamfer sums.
// Output: [total.mean, chamfer.mean, transform.mean, transform_chamfer.mean]
// ---------------------------------------------------------------------------
__global__ void finalize_kernel(const float* __restrict__ acc,
                                const float* __restrict__ predT,
                                const float* __restrict__ gtT,
                                float* __restrict__ out) {
    if (threadIdx.x != 0 || blockIdx.x != 0) return;
    const float invN = 1.0f / (float)NPTS;
    float tot = 0.0f, chm = 0.0f, trf = 0.0f, tcm = 0.0f;
    for (int b = 0; b < BATCH; ++b) {
        float cxt = (acc[0 * BATCH + b] + acc[1 * BATCH + b]) * invN;
        float cxg = (acc[2 * BATCH + b] + acc[3 * BATCH + b]) * invN;
        const float* P = predT + b * 16;
        const float* G = gtT + b * 16;
        float r2 = 0.0f;
        for (int j = 0; j < 3; ++j)
            for (int k = 0; k < 3; ++k) {
                float d = P[j * 4 + k] - G[j * 4 + k];
                r2 += d * d;
            }
        float t2 = 0.0f;
        for (int j = 0; j < 3; ++j) {
            float d = P[j * 4 + 3] - G[j * 4 + 3];
            t2 += d * d;
        }
        float tl = sqrtf(r2) + sqrtf(t2);
        tot += cxt + tl + 0.5f * cxg;
        chm += cxt;
        trf += tl;
        tcm += cxg;
    }
    const float invB = 1.0f / (float)BATCH;
    out[0] = tot * invB;
    out[1] = chm * invB;
    out[2] = trf * invB;
    out[3] = tcm * invB;
}

// ---------------------------------------------------------------------------
extern "C" void kernel_launch(void* const* d_in, const int* in_sizes, int n_in,
                              void* d_out, int out_size, void* d_ws, size_t ws_size,
                              hipStream_t stream) {
    const float* predT = (const float*)d_in[0];   // (8,4,4)
    const float* gtT   = (const float*)d_in[1];   // (8,4,4)
    const float* src   = (const float*)d_in[2];   // (8,4096,3)
    const float* tgt   = (const float*)d_in[3];   // (8,4096,3)

    char* ws = (char*)d_ws;
    const size_t PACK_BYTES = (size_t)BATCH * NPTS * sizeof(float4);  // 512 KB each
    float*  acc      = (float*)ws;                                    // 32 floats
    float4* packPred = (float4*)(ws + 256);
    float4* packTgt  = (float4*)(ws + 256 + PACK_BYTES);
    float4* packGt   = (float4*)(ws + 256 + 2 * PACK_BYTES);

    zero_acc_kernel<<<1, 32, 0, stream>>>(acc);

    int nprep = BATCH * NPTS;
    prep_points_kernel<<<(nprep + 255) / 256, 256, 0, stream>>>(
        predT, gtT, src, tgt, packPred, packTgt, packGt);

    int total_threads = NPROB * BATCH * TILES * 32;   // 8192 waves
    chamfer_rowmin_kernel<<<total_threads / 256, 256, 0, stream>>>(
        packPred, packTgt, packGt, acc);

    finalize_kernel<<<1, 1, 0, stream>>>(acc, predT, gtT, (float*)d_out);
}